// Dial2vec_5377299054888
// MI455X (gfx1250) — compile-verified
//
#include <hip/hip_runtime.h>
#include <math.h>
#include <stdint.h>

// ---- problem constants (match reference) ----
#define BSEQ 80
#define L 512
#define H 768
#define SAMPLES 10
#define VIEW_RANGE 2
#define TEMP 0.2f
#define AVG_EPS 1e-6f
#define COS_EPS 1e-8f

// ---- tiling ----
#define KCH 8           // K chunk streamed through LDS (double buffered)
#define LSTR 12         // LDS row stride in floats (16B aligned, conflict-free)
#define NCHUNK (H / KCH)
#define JBLK 64         // rows of w per workgroup
#define NWAVES 8        // 256 threads, wave32

typedef __attribute__((ext_vector_type(2))) float v2f;
typedef __attribute__((ext_vector_type(8))) float v8f;

// ---------------------------------------------------------------------
// Async global->LDS copy (CDNA5 GLOBAL_LOAD_ASYNC_TO_LDS_B128, ASYNCcnt)
// with synchronous fallback if the builtin is unavailable.
// Probe-derived signature: param0 is AS1 (global) int4*, per hipcc
// diagnostic "__vector_size__(4*sizeof(int)) int __device__ *".
// ---------------------------------------------------------------------
#if defined(__has_builtin)
#if __has_builtin(__builtin_amdgcn_global_load_async_to_lds_b128)
#define D2V_HAS_ASYNC 1
#endif
#endif

typedef int d2v_v4i __attribute__((vector_size(16)));
typedef __attribute__((address_space(1))) d2v_v4i* d2v_gvp;
typedef __attribute__((address_space(3))) d2v_v4i* d2v_lvp;

__device__ __forceinline__ void d2v_cp16(const float* g, float* l) {
#ifdef D2V_HAS_ASYNC
    __builtin_amdgcn_global_load_async_to_lds_b128(
        (d2v_gvp)(uintptr_t)g,
        (d2v_lvp)(unsigned)(uintptr_t)l,   // low 32 bits == LDS byte offset
        0 /*imm offset*/, 0 /*cpol*/);
#else
    *(float4*)l = *(const float4*)g;
#endif
}
__device__ __forceinline__ void d2v_wait_group() {   // allow 4 in flight
#ifdef D2V_HAS_ASYNC
    asm volatile("s_wait_asynccnt 0x4" ::: "memory");
#endif
}
__device__ __forceinline__ void d2v_wait_all() {
#ifdef D2V_HAS_ASYNC
    asm volatile("s_wait_asynccnt 0x0" ::: "memory");
#endif
}

// =====================================================================
// Kernel 1: fp32 WMMA GEMM  w = (h h^T) tile, fused band/role masking,
// produces alpha'[b][512] (row sums) and beta' partials [b][8][512].
// Double-buffered async K-streaming: DMA chunk c+1 overlaps WMMA on c.
// =====================================================================
__global__ __launch_bounds__(256) void d2v_gemm_band(
    const float* __restrict__ hidden,
    const int* __restrict__ attn,
    const int* __restrict__ role,
    const int* __restrict__ turn,
    float* __restrict__ beta_part,   // [80][8][512]
    float* __restrict__ alpha_out)   // [80][512]
{
    __shared__ float sh_t[2][L * LSTR];         // 2 x 24 KB double buffer
    __shared__ float sh_am[L];
    __shared__ float sh_bm[L];
    __shared__ int   sh_turn[L];
    __shared__ float sh_alpha_w[NWAVES][JBLK];  // per-wave alpha partials

    const int tid  = threadIdx.x;
    const int b    = blockIdx.x >> 3;
    const int jb   = blockIdx.x & 7;
    const int j0   = jb * JBLK;

    const int lane = tid & 31;
    const int wv   = tid >> 5;
    const int n_in = lane & 15;
    const int hf   = lane >> 4;       // half-wave (K/M split in WMMA layouts)

    const float* hb = hidden + (size_t)b * L * H;

    // per-thread quad assignment: 512 rows x 8 floats = 1024 float4 / chunk
    int rrow[4], rqi[4];
    #pragma unroll
    for (int p = 0; p < 4; ++p) {
        int q = p * 256 + tid;
        rrow[p] = q >> 1;
        rqi[p]  = (q & 1) * 4;
    }

    // masks + turn ids for this batch element
    for (int p = 0; p < 2; ++p) {
        int i = tid + p * 256;
        int a  = attn[b * L + i];
        int ro = role[b * L + i];
        sh_am[i]   = (float)(a * (ro == 0));
        sh_bm[i]   = (float)(a * (ro == 1));
        sh_turn[i] = turn[b * L + i];
    }

    v8f acc[4][4];
    #pragma unroll
    for (int tm = 0; tm < 4; ++tm)
        #pragma unroll
        for (int tn = 0; tn < 4; ++tn)
            acc[tm][tn] = (v8f){0.f,0.f,0.f,0.f,0.f,0.f,0.f,0.f};

    // prologue: DMA chunk 0 into buffer 0
    #pragma unroll
    for (int p = 0; p < 4; ++p)
        d2v_cp16(hb + rrow[p] * H + rqi[p], &sh_t[0][rrow[p] * LSTR + rqi[p]]);

    for (int c = 0; c < NCHUNK; ++c) {
        float* cur = sh_t[c & 1];
        float* nxt = sh_t[(c & 1) ^ 1];
        if (c + 1 < NCHUNK) {
            const int kb = (c + 1) * KCH;
            #pragma unroll
            for (int p = 0; p < 4; ++p)
                d2v_cp16(hb + rrow[p] * H + kb + rqi[p],
                         &nxt[rrow[p] * LSTR + rqi[p]]);
            d2v_wait_group();      // chunk c complete; next chunk in flight
        } else {
            d2v_wait_all();
        }
        __syncthreads();           // all waves' chunk-c data visible

        #pragma unroll
        for (int ks = 0; ks < KCH / 4; ++ks) {
            const int ko = 4 * ks + 2 * hf;   // lanes 0-15: K+0/+1, 16-31: K+2/+3
            v2f aF[4], bF[4];
            #pragma unroll
            for (int tm = 0; tm < 4; ++tm)    // A: rows of this block's j-slab
                aF[tm] = *(const v2f*)(cur + (j0 + tm * 16 + n_in) * LSTR + ko);
            #pragma unroll
            for (int tn = 0; tn < 4; ++tn)    // B: rows of this wave's i-slab
                bF[tn] = *(const v2f*)(cur + (wv * 64 + tn * 16 + n_in) * LSTR + ko);
            #pragma unroll
            for (int tm = 0; tm < 4; ++tm)
                #pragma unroll
                for (int tn = 0; tn < 4; ++tn)
                    acc[tm][tn] = __builtin_amdgcn_wmma_f32_16x16x4_f32(
                        false, aF[tm], false, bF[tn],
                        (short)0, acc[tm][tn], false, false);
        }
        __syncthreads();           // done reading cur; it may be overwritten
    }

    // ---- epilogue: band mask + row/col reductions (deterministic) ----
    float bmi[4]; int ti[4];
    #pragma unroll
    for (int tn = 0; tn < 4; ++tn) {
        int i = wv * 64 + tn * 16 + n_in;
        bmi[tn] = sh_bm[i];
        ti[tn]  = sh_turn[i];
    }
    float beta_acc[4] = {0.f, 0.f, 0.f, 0.f};

    #pragma unroll
    for (int tm = 0; tm < 4; ++tm) {
        #pragma unroll
        for (int e = 0; e < 8; ++e) {
            int j   = tm * 16 + e + 8 * hf;      // local row 0..63
            int tj  = sh_turn[j0 + j];
            float amj = sh_am[j0 + j];
            float ap = 0.f;
            #pragma unroll
            for (int tn = 0; tn < 4; ++tn) {
                int dd = tj - ti[tn];
                float band = (dd <= VIEW_RANGE && dd >= -VIEW_RANGE) ? 1.f : 0.f;
                float wval = acc[tm][tn][e];
                ap           += band * bmi[tn] * wval;  // alpha'[j]
                beta_acc[tn] += band * amj    * wval;   // beta'[i]
            }
            // reduce alpha over the 16 lanes of this half
            ap += __shfl_xor(ap, 1, 32);
            ap += __shfl_xor(ap, 2, 32);
            ap += __shfl_xor(ap, 4, 32);
            ap += __shfl_xor(ap, 8, 32);
            if (n_in == 0) sh_alpha_w[wv][j] = ap;
        }
    }
    // beta: combine the two halves (same i), waves own disjoint i-slabs
    #pragma unroll
    for (int tn = 0; tn < 4; ++tn) {
        float bsum = beta_acc[tn] + __shfl_xor(beta_acc[tn], 16, 32);
        if (hf == 0)
            beta_part[(b * 8 + jb) * L + wv * 64 + tn * 16 + n_in] = bsum;
    }

    __syncthreads();
    if (tid < JBLK) {
        float s = 0.f;
        #pragma unroll
        for (int w = 0; w < NWAVES; ++w) s += sh_alpha_w[w][tid];
        alpha_out[b * L + j0 + tid] = s;
    }
}

// =====================================================================
// Kernel 2: per batch element, one pass over hidden -> qs/qc/rs/rc
// vectors, cosine logits.
// =====================================================================
__global__ __launch_bounds__(256) void d2v_vec(
    const float* __restrict__ hidden,
    const int* __restrict__ attn,
    const int* __restrict__ role,
    const float* __restrict__ beta_part,
    const float* __restrict__ alpha,
    float* __restrict__ logit_q,
    float* __restrict__ logit_r)
{
    __shared__ float s_am[L], s_bm[L], s_al[L], s_be[L];
    __shared__ float red[6][256];

    const int b = blockIdx.x;
    const int tid = threadIdx.x;
    const float* hb = hidden + (size_t)b * L * H;

    for (int p = 0; p < 2; ++p) {
        int i = tid + p * 256;
        int a  = attn[b * L + i];
        int ro = role[b * L + i];
        s_am[i] = (float)(a * (ro == 0));
        s_bm[i] = (float)(a * (ro == 1));
        s_al[i] = alpha[b * L + i];
        float s = 0.f;
        #pragma unroll
        for (int jb = 0; jb < 8; ++jb) s += beta_part[(b * 8 + jb) * L + i];
        s_be[i] = s;
    }
    __syncthreads();

    // token counts (deterministic tree)
    red[0][tid] = s_am[tid] + s_am[tid + 256];
    red[1][tid] = s_bm[tid] + s_bm[tid + 256];
    __syncthreads();
    for (int off = 128; off > 0; off >>= 1) {
        if (tid < off) { red[0][tid] += red[0][tid + off];
                         red[1][tid] += red[1][tid + off]; }
        __syncthreads();
    }
    const float cntA = red[0][0], cntB = red[1][0];
    __syncthreads();

    float qs[3] = {0,0,0}, qc[3] = {0,0,0}, rs[3] = {0,0,0}, rc[3] = {0,0,0};
    for (int i = 0; i < L; ++i) {
        float am = s_am[i], bm = s_bm[i];
        float wqc = am * s_al[i];
        float wrc = bm * s_be[i];
        const float* hrow = hb + i * H;
        #pragma unroll
        for (int k = 0; k < 3; ++k) {
            float h = hrow[tid + 256 * k];
            qs[k] += am  * h;  qc[k] += wqc * h;
            rs[k] += bm  * h;  rc[k] += wrc * h;
        }
    }

    const float iA = 1.f / (cntA + AVG_EPS);
    const float iB = 1.f / (cntB + AVG_EPS);
    float dq = 0, nqs = 0, nqc = 0, dr = 0, nrs = 0, nrc = 0;
    #pragma unroll
    for (int k = 0; k < 3; ++k) {
        float a = qs[k] * iA, c = qc[k] * iB;   // masked averages
        float x = rs[k] * iB, y = rc[k] * iA;
        dq += a * c; nqs += a * a; nqc += c * c;
        dr += x * y; nrs += x * x; nrc += y * y;
    }
    red[0][tid] = dq;  red[1][tid] = nqs; red[2][tid] = nqc;
    red[3][tid] = dr;  red[4][tid] = nrs; red[5][tid] = nrc;
    __syncthreads();
    for (int off = 128; off > 0; off >>= 1) {
        if (tid < off)
            #pragma unroll
            for (int j = 0; j < 6; ++j) red[j][tid] += red[j][tid + off];
        __syncthreads();
    }
    if (tid == 0) {
        float nx = fmaxf(sqrtf(red[1][0]), COS_EPS);
        float ny = fmaxf(sqrtf(red[2][0]), COS_EPS);
        logit_q[b] = (red[0][0] / (nx * ny)) * (1.f / TEMP);
        nx = fmaxf(sqrtf(red[4][0]), COS_EPS);
        ny = fmaxf(sqrtf(red[5][0]), COS_EPS);
        logit_r[b] = (red[3][0] / (nx * ny)) * (1.f / TEMP);
    }
}

// =====================================================================
// Kernel 3: log-softmax losses -> scalar
// =====================================================================
__global__ void d2v_loss(const float* __restrict__ labels,
                         const float* __restrict__ lq,
                         const float* __restrict__ lr,
                         float* __restrict__ out)
{
    __shared__ float sq[8], sr[8];
    int tid = threadIdx.x;
    if (tid < 8) {
        float accq = 0.f, accr = 0.f;
        {
            float m = -1e30f;
            for (int j = 0; j < SAMPLES; ++j) m = fmaxf(m, lq[tid * SAMPLES + j]);
            float s = 0.f;
            for (int j = 0; j < SAMPLES; ++j) s += expf(lq[tid * SAMPLES + j] - m);
            float lse = m + logf(s);
            for (int j = 0; j < SAMPLES; ++j)
                accq += (lq[tid * SAMPLES + j] - lse) * labels[tid * SAMPLES + j];
        }
        {
            float m = -1e30f;
            for (int j = 0; j < SAMPLES; ++j) m = fmaxf(m, lr[tid * SAMPLES + j]);
            float s = 0.f;
            for (int j = 0; j < SAMPLES; ++j) s += expf(lr[tid * SAMPLES + j] - m);
            float lse = m + logf(s);
            for (int j = 0; j < SAMPLES; ++j)
                accr += (lr[tid * SAMPLES + j] - lse) * labels[tid * SAMPLES + j];
        }
        sq[tid] = accq; sr[tid] = accr;
    }
    __syncthreads();
    if (tid == 0) {
        float a = 0.f, c = 0.f;
        for (int i = 0; i < 8; ++i) { a += sq[i]; c += sr[i]; }
        out[0] = -(a + c) / (float)BSEQ;   // loss_q + loss_r, each -mean over 8x10
    }
}

// =====================================================================
extern "C" void kernel_launch(void* const* d_in, const int* in_sizes, int n_in,
                              void* d_out, int out_size, void* d_ws, size_t ws_size,
                              hipStream_t stream) {
    const float* hidden = (const float*)d_in[0];
    const float* labels = (const float*)d_in[1];
    const int*   attn   = (const int*)d_in[2];
    const int*   role   = (const int*)d_in[3];
    const int*   turn   = (const int*)d_in[4];

    float* ws        = (float*)d_ws;
    float* beta_part = ws;                              // 80*8*512
    float* alpha     = beta_part + BSEQ * 8 * L;        // 80*512
    float* lq        = alpha + BSEQ * L;                // 80
    float* lr        = lq + BSEQ;                       // 80

    d2v_gemm_band<<<BSEQ * 8, 256, 0, stream>>>(hidden, attn, role, turn,
                                                beta_part, alpha);
    d2v_vec<<<BSEQ, 256, 0, stream>>>(hidden, attn, role, beta_part, alpha, lq, lr);
    d2v_loss<<<1, 64, 0, stream>>>(labels, lq, lr, (float*)d_out);
}